// Model_21689584844835
// MI455X (gfx1250) — compile-verified
//
#include <hip/hip_runtime.h>
#include <math.h>

#define N_NODES 150000
#define N_EDGES 1200000
#define NPG     30
#define NG      5000

typedef float v2f __attribute__((ext_vector_type(2)));
typedef float v8f __attribute__((ext_vector_type(8)));

// ---------- helpers ----------
__device__ inline unsigned f2ord(float f) {
    unsigned u = __float_as_uint(f);
    return (u & 0x80000000u) ? ~u : (u | 0x80000000u);
}
__device__ inline float ord2f(unsigned k) {
    unsigned u = (k & 0x80000000u) ? (k ^ 0x80000000u) : ~k;
    return __uint_as_float(u);
}
__device__ inline float wave_sum(float v) {
    for (int m = 16; m > 0; m >>= 1) v += __shfl_xor(v, m, 32);
    return v;
}
__device__ inline float wave_max(float v) {
    for (int m = 16; m > 0; m >>= 1) v = fmaxf(v, __shfl_xor(v, m, 32));
    return v;
}
__device__ inline float lrelu(float v) { return v > 0.f ? v : 0.2f * v; }

// ---------- init ----------
__global__ void k_fill(float* p, float v, long n) {
    long i = (long)blockIdx.x * blockDim.x + threadIdx.x;
    if (i < n) p[i] = v;
}
__global__ void k_init_red(float* red) {
    red[0] = red[1] = red[2] = 0.f;               // column sums
    unsigned* u = (unsigned*)red;
    unsigned ninf = f2ord(-INFINITY);
    u[3] = u[4] = u[5] = ninf;                    // column max keys
    u[6] = ninf;                                  // softmax max key
    red[7] = 0.f;                                 // softmax sum
}

// ---------- input normalization ----------
__global__ void k_colstats(const float* __restrict__ x, float* red) {
    int i = blockIdx.x * blockDim.x + threadIdx.x;
    float s0 = 0.f, s1 = 0.f, s2 = 0.f;
    float m0 = -INFINITY, m1 = -INFINITY, m2 = -INFINITY;
    if (i < N_NODES) {
        s0 = x[3 * i + 0]; s1 = x[3 * i + 1]; s2 = x[3 * i + 2];
        m0 = s0; m1 = s1; m2 = s2;
    }
    s0 = wave_sum(s0); s1 = wave_sum(s1); s2 = wave_sum(s2);
    m0 = wave_max(m0); m1 = wave_max(m1); m2 = wave_max(m2);
    if ((threadIdx.x & 31) == 0) {
        atomicAdd(&red[0], s0); atomicAdd(&red[1], s1); atomicAdd(&red[2], s2);
        unsigned* u = (unsigned*)red;
        atomicMax(&u[3], f2ord(m0)); atomicMax(&u[4], f2ord(m1)); atomicMax(&u[5], f2ord(m2));
    }
}
__global__ void k_normalize(const float* __restrict__ x, const float* red, float* x0) {
    long idx = (long)blockIdx.x * blockDim.x + threadIdx.x;
    if (idx >= (long)N_NODES * 3) return;
    int c = (int)(idx % 3);
    float mean = red[c] * (1.0f / N_NODES);
    float mx = ord2f(((const unsigned*)red)[3 + c]);
    x0[idx] = (x[idx] - mean) / (mx - mean);      // max of centered col = max - mean
}

// ---------- edge scatter: both GCN (weighted) and GIN (raw) aggregates ----------
__global__ void k_edge_scatter(const int* __restrict__ row, const int* __restrict__ col,
                               const float* __restrict__ ew, const float* __restrict__ h,
                               int din, float* aggGcn, float* aggGin, long total) {
    long idx = (long)blockIdx.x * blockDim.x + threadIdx.x;
    if (idx >= total) return;
    int e = (int)(idx / din);
    int f = (int)(idx - (long)e * din);
    int r = row[e], c = col[e];
    float v = h[(size_t)r * din + f];
    atomicAdd(&aggGin[(size_t)c * din + f], v);
    atomicAdd(&aggGcn[(size_t)c * din + f], v * ew[e]);
}

// ---------- WMMA f32 GEMM with fused bias: C[:, :dout] = A@B + bias ----------
// A: [M, K] lda=K   B: [K, 16*gridDim.y] ldb   C row stride ldc
__global__ void k_wmma_gemm_bias(const float* __restrict__ A, int lda,
                                 const float* __restrict__ B, int ldb,
                                 const float* __restrict__ bias,
                                 float* __restrict__ C, int ldc, int M, int K) {
    int wave = threadIdx.x >> 5;
    int lane = threadIdx.x & 31;
    int mtile = blockIdx.x * 4 + wave;
    if (mtile * 16 >= M) return;                  // wave-uniform exit
    int m0 = mtile * 16;
    int col = (blockIdx.y << 4) + (lane & 15);
    int arow = m0 + (lane & 15);
    int khalf = (lane >> 4) * 2;                  // 0 or 2
    float bval = bias[col];
    v8f c;
    #pragma unroll
    for (int j = 0; j < 8; j++) c[j] = bval;
    for (int k = 0; k < K; k += 4) {
        v2f a, b;
        a[0] = A[(size_t)arow * lda + k + khalf];
        a[1] = A[(size_t)arow * lda + k + khalf + 1];
        b[0] = B[(size_t)(k + khalf) * ldb + col];
        b[1] = B[(size_t)(k + khalf + 1) * ldb + col];
        c = __builtin_amdgcn_wmma_f32_16x16x4_f32(false, a, false, b, (short)0, c,
                                                  false, false);
    }
    int crow = m0 + (lane >> 4) * 8;
    #pragma unroll
    for (int j = 0; j < 8; j++)
        C[(size_t)(crow + j) * ldc + col] = c[j];
}

// ---------- layer-1 GCN GEMM (K=3, dout=32) ----------
__global__ void k_gemm3(const float* __restrict__ agg, const float* __restrict__ W,
                        const float* __restrict__ b, float* __restrict__ xout) {
    int idx = blockIdx.x * blockDim.x + threadIdx.x;
    int i = idx >> 5, j = idx & 31;
    if (i >= N_NODES) return;
    float s = b[j];
    #pragma unroll
    for (int k = 0; k < 3; k++) s += agg[i * 3 + k] * W[k * 32 + j];
    xout[(size_t)i * 64 + j] = s;
}

// ---------- leaky relu on GCN half (in place) ----------
__global__ void k_lrelu(float* xout, int d, int ld, long n) {
    long idx = (long)blockIdx.x * blockDim.x + threadIdx.x;
    if (idx >= n) return;
    long i = idx / d;
    int f = (int)(idx - i * d);
    float* p = xout + (size_t)i * ld + f;
    *p = lrelu(*p);
}

// ---------- GIN MLP: relu(relu(((1+eps)h+agg)@W1+b1)@W2+b2) -> xout[:, d:2d] ----------
__global__ void k_gin(const float* __restrict__ h, const float* __restrict__ agg,
                      int din, int d,
                      const float* __restrict__ W1, const float* __restrict__ b1,
                      const float* __restrict__ W2, const float* __restrict__ b2,
                      const float* __restrict__ eps, float* __restrict__ xout, int ld) {
    int i = blockIdx.x * blockDim.x + threadIdx.x;
    if (i >= N_NODES) return;
    const float* hr = h + (size_t)i * din;
    const float* ar = agg + (size_t)i * din;
    float ep = 1.0f + eps[0];
    float u[10];
    #pragma unroll
    for (int j = 0; j < 10; j++) u[j] = b1[j];
    for (int f = 0; f < din; f++) {
        float m = ep * hr[f] + ar[f];
        #pragma unroll
        for (int j = 0; j < 10; j++) u[j] += m * W1[f * 10 + j];
    }
    #pragma unroll
    for (int j = 0; j < 10; j++) u[j] = fmaxf(u[j], 0.f);
    float* orow = xout + (size_t)i * ld + d;
    for (int g = 0; g < d; g++) {
        float s = b2[g];
        #pragma unroll
        for (int j = 0; j < 10; j++) s += u[j] * W2[j * d + g];
        orow[g] = fmaxf(s, 0.f);
    }
}

// ---------- GAT ----------
__global__ void k_hj(const float* __restrict__ x3, const float* __restrict__ gw,
                     float* __restrict__ hj) {
    int i = blockIdx.x * blockDim.x + threadIdx.x;
    if (i >= N_NODES) return;
    const float* xr = x3 + (size_t)i * 256;
    float s = 0.f;
    for (int f = 0; f < 256; f++) s += xr[f] * gw[f];
    hj[i] = s;
}
__global__ void k_edge_max(const int* __restrict__ row, const float* __restrict__ hj,
                           const float* __restrict__ att, float* red) {
    int e = blockIdx.x * blockDim.x + threadIdx.x;
    float a = att[0];
    float z = -INFINITY;
    if (e < N_EDGES) z = lrelu(hj[row[e]] * a);
    z = wave_max(z);
    if ((threadIdx.x & 31) == 0) atomicMax(&((unsigned*)red)[6], f2ord(z));
}
__global__ void k_edge_sumexp(const int* __restrict__ row, const float* __restrict__ hj,
                              const float* __restrict__ att, float* red) {
    int e = blockIdx.x * blockDim.x + threadIdx.x;
    float a = att[0];
    float m = ord2f(((const unsigned*)red)[6]);
    float p = 0.f;
    if (e < N_EDGES) p = expf(lrelu(hj[row[e]] * a) - m);
    p = wave_sum(p);
    if ((threadIdx.x & 31) == 0) atomicAdd(&red[7], p);
}
__global__ void k_edge_att(const int* __restrict__ row, const int* __restrict__ col,
                           const float* __restrict__ hj, const float* __restrict__ att,
                           const float* red, float* attw) {
    int e = blockIdx.x * blockDim.x + threadIdx.x;
    if (e >= N_EDGES) return;
    float a = att[0];
    float m = ord2f(((const unsigned*)red)[6]);
    float inv = 1.0f / red[7];
    float hv = hj[row[e]];
    float w = hv * expf(lrelu(hv * a) - m) * inv;
    atomicAdd(&attw[col[e]], w);
}

// ---------- pool + readout: one wave32 per graph ----------
__global__ void k_readout(const float* __restrict__ x3, const float* __restrict__ attw,
                          const float* __restrict__ lw, const float* __restrict__ lb,
                          const float* __restrict__ l2w, const float* __restrict__ l2b,
                          float* __restrict__ out) {
    int wid = ((blockIdx.x * blockDim.x + threadIdx.x) >> 5);
    int lane = threadIdx.x & 31;
    if (wid >= NG) return;
    float p[8];
    #pragma unroll
    for (int k = 0; k < 8; k++) p[k] = 0.f;
    int nbase = wid * NPG;
    for (int n = 0; n < NPG; n++) {
        float w = attw[nbase + n];
        const float* xr = x3 + (size_t)(nbase + n) * 256;
        #pragma unroll
        for (int k = 0; k < 8; k++) p[k] += xr[lane + 32 * k] * w;
    }
    #pragma unroll
    for (int k = 0; k < 8; k++) p[k] *= (1.0f / NPG);
    float total = l2b[0];
    for (int j = 0; j < 10; j++) {
        float part = 0.f;
        #pragma unroll
        for (int k = 0; k < 8; k++) part += p[k] * lw[(lane + 32 * k) * 10 + j];
        part = wave_sum(part);
        total += (part + lb[j]) * l2w[j];
    }
    if (lane == 0) out[wid] = 1.0f / (1.0f + expf(-total));
}

// =================== host ===================
static inline unsigned gblocks(long n, int b) { return (unsigned)((n + b - 1) / b); }

extern "C" void kernel_launch(void* const* d_in, const int* in_sizes, int n_in,
                              void* d_out, int out_size, void* d_ws, size_t ws_size,
                              hipStream_t stream) {
    const float* x       = (const float*)d_in[0];
    const int*   ei      = (const int*)  d_in[1];
    const float* ew      = (const float*)d_in[2];
    const float* gcn1_w  = (const float*)d_in[3];  const float* gcn1_b = (const float*)d_in[4];
    const float* gcn2_w  = (const float*)d_in[5];  const float* gcn2_b = (const float*)d_in[6];
    const float* gcn3_w  = (const float*)d_in[7];  const float* gcn3_b = (const float*)d_in[8];
    const float* gin1_w1 = (const float*)d_in[9];  const float* gin1_b1 = (const float*)d_in[10];
    const float* gin1_w2 = (const float*)d_in[11]; const float* gin1_b2 = (const float*)d_in[12];
    const float* gin1_e  = (const float*)d_in[13];
    const float* gin2_w1 = (const float*)d_in[14]; const float* gin2_b1 = (const float*)d_in[15];
    const float* gin2_w2 = (const float*)d_in[16]; const float* gin2_b2 = (const float*)d_in[17];
    const float* gin2_e  = (const float*)d_in[18];
    const float* gin3_w1 = (const float*)d_in[19]; const float* gin3_b1 = (const float*)d_in[20];
    const float* gin3_w2 = (const float*)d_in[21]; const float* gin3_b2 = (const float*)d_in[22];
    const float* gin3_e  = (const float*)d_in[23];
    const float* gat_w   = (const float*)d_in[24]; const float* gat_att = (const float*)d_in[25];
    const float* lin_w   = (const float*)d_in[26]; const float* lin_b   = (const float*)d_in[27];
    const float* lin2_w  = (const float*)d_in[28]; const float* lin2_b  = (const float*)d_in[29];
    const int* row = ei;
    const int* col = ei + N_EDGES;
    float* out = (float*)d_out;

    // workspace layout (all f32)
    float* ws = (float*)d_ws;
    size_t o = 0;
    float* x0   = ws + o; o += (size_t)N_NODES * 3;
    float* x1   = ws + o; o += (size_t)N_NODES * 64;
    float* x2   = ws + o; o += (size_t)N_NODES * 128;
    float* x3   = ws + o; o += (size_t)N_NODES * 256;
    float* aggG = ws + o; o += (size_t)N_NODES * 128;   // GCN aggregate (max din)
    float* aggI = ws + o; o += (size_t)N_NODES * 128;   // GIN aggregate (max din)
    float* hj   = ws + o; o += (size_t)N_NODES;
    float* attw = ws + o; o += (size_t)N_NODES;
    float* red  = ws + o; o += 16;
    (void)ws_size; (void)n_in; (void)in_sizes; (void)out_size;

    const int B = 256;

    // ---- input normalization ----
    k_init_red<<<1, 1, 0, stream>>>(red);
    k_colstats<<<gblocks(N_NODES, B), B, 0, stream>>>(x, red);
    k_normalize<<<gblocks((long)N_NODES * 3, B), B, 0, stream>>>(x, red, x0);

    // ---- generic layer runner ----
    auto layer = [&](const float* h, int din, int d,
                     const float* Wg, const float* bg,
                     const float* W1, const float* b1,
                     const float* W2, const float* b2,
                     const float* eps, float* xout, bool use_wmma) {
        long na = (long)N_NODES * din;
        k_fill<<<gblocks(na, B), B, 0, stream>>>(aggG, 0.f, na);
        k_fill<<<gblocks(na, B), B, 0, stream>>>(aggI, 0.f, na);
        long tot = (long)N_EDGES * din;
        k_edge_scatter<<<gblocks(tot, B), B, 0, stream>>>(row, col, ew, h, din, aggG, aggI, tot);
        int ld = 2 * d;
        if (use_wmma) {
            dim3 grid((N_NODES / 16 + 3) / 4, d / 16);
            k_wmma_gemm_bias<<<grid, 128, 0, stream>>>(aggG, din, Wg, d, bg, xout, ld,
                                                       N_NODES, din);
        } else {
            k_gemm3<<<gblocks((long)N_NODES * 32, B), B, 0, stream>>>(aggG, Wg, bg, xout);
        }
        k_lrelu<<<gblocks((long)N_NODES * d, B), B, 0, stream>>>(xout, d, ld, (long)N_NODES * d);
        k_gin<<<gblocks(N_NODES, B), B, 0, stream>>>(h, aggI, din, d, W1, b1, W2, b2, eps,
                                                     xout, ld);
    };

    layer(x0, 3,   32,  gcn1_w, gcn1_b, gin1_w1, gin1_b1, gin1_w2, gin1_b2, gin1_e, x1, false);
    layer(x1, 64,  64,  gcn2_w, gcn2_b, gin2_w1, gin2_b1, gin2_w2, gin2_b2, gin2_e, x2, true);
    layer(x2, 128, 128, gcn3_w, gcn3_b, gin3_w1, gin3_b1, gin3_w2, gin3_b2, gin3_e, x3, true);

    // ---- GAT (global softmax over all edges) ----
    k_hj<<<gblocks(N_NODES, B), B, 0, stream>>>(x3, gat_w, hj);
    k_edge_max<<<gblocks(N_EDGES, B), B, 0, stream>>>(row, hj, gat_att, red);
    k_edge_sumexp<<<gblocks(N_EDGES, B), B, 0, stream>>>(row, hj, gat_att, red);
    k_fill<<<gblocks(N_NODES, B), B, 0, stream>>>(attw, 0.f, N_NODES);
    k_edge_att<<<gblocks(N_EDGES, B), B, 0, stream>>>(row, col, hj, gat_att, red, attw);

    // ---- pool + readout ----
    k_readout<<<gblocks((long)NG * 32, B), B, 0, stream>>>(x3, attw, lin_w, lin_b,
                                                           lin2_w, lin2_b, out);
}